// CustomStellarEncoder_0_7000796693090
// MI455X (gfx1250) — compile-verified
//
#include <hip/hip_runtime.h>

#define N_NODES 100000
#define N_EDGES 1600000
#define IN_DIM  48
#define HID     128
#define BN_EPS  1e-5f

typedef __attribute__((ext_vector_type(2))) float v2f;
typedef __attribute__((ext_vector_type(8))) float v8f;

// ---------------------------------------------------------------------------
// GEMM1: h = x @ W1^T + b1     x:[N,48]  W1:[128,48]  h:[N,128]
// One workgroup = 128 rows. 8 waves; wave w owns M-tile rows [16w,16w+16).
// WMMA f32 16x16x4: A lane layout: lane l -> M=l%16, K=2*(l/16)+{0,1}
// ---------------------------------------------------------------------------
template <bool FULL>
__device__ __forceinline__ void gemm1_body(const float* __restrict__ xs,
                                           const float* __restrict__ ws, int LDK,
                                           const float* __restrict__ b1,
                                           float* __restrict__ h, int m_base,
                                           int lane, int wave) {
  const int m0 = wave * 16;
  const int lm = lane & 15;
  const int lk = (lane >> 4) << 1;
  const float* zrow = &xs[(m0 + lm) * LDK + lk];
  const int rbase = m_base + m0 + ((lane >> 4) << 3);

  // preload all biases so the WMMA loop contains no VMEM ops
  float biasv[8];
#pragma unroll
  for (int t = 0; t < 8; ++t) biasv[t] = b1[t * 16 + lm];

#pragma unroll
  for (int t = 0; t < 8; ++t) {
    const int n0 = t * 16;
    v8f acc = {};
    const float* wrow = &ws[(n0 + lm) * LDK + lk];
#pragma unroll
    for (int k0 = 0; k0 < IN_DIM; k0 += 4) {
      v2f a, b;
      a.x = zrow[k0];  a.y = zrow[k0 + 1];
      b.x = wrow[k0];  b.y = wrow[k0 + 1];
      acc = __builtin_amdgcn_wmma_f32_16x16x4_f32(false, a, false, b, (short)0, acc,
                                                  false, false);
    }
    const int col = n0 + lm;
    const float bias = biasv[t];
    if (FULL) {
#pragma unroll
      for (int v = 0; v < 8; ++v)
        h[(size_t)(rbase + v) * HID + col] = acc[v] + bias;
    } else {
#pragma unroll
      for (int v = 0; v < 8; ++v) {
        int row = rbase + v;
        if (row < N_NODES) h[(size_t)row * HID + col] = acc[v] + bias;
      }
    }
  }
}

__global__ __launch_bounds__(256) void gemm1_kernel(const float* __restrict__ x,
                                                    const float* __restrict__ W1,
                                                    const float* __restrict__ b1,
                                                    float* __restrict__ h) {
  constexpr int LDK = IN_DIM + 3;  // 51: odd stride -> conflict-free frag reads
  __shared__ float xs[128 * LDK];
  __shared__ float ws[HID * LDK];
  const int tid = threadIdx.x;
  const int m_base = blockIdx.x * 128;
  const bool full = (m_base + 128 <= N_NODES);  // uniform per workgroup

  for (int i = tid; i < HID * IN_DIM; i += 256)
    ws[(i / IN_DIM) * LDK + (i % IN_DIM)] = W1[i];
  for (int i = tid; i < 128 * IN_DIM; i += 256) {
    int m = i / IN_DIM, k = i % IN_DIM;
    int gm = m_base + m;
    xs[m * LDK + k] = (gm < N_NODES) ? x[(size_t)gm * IN_DIM + k] : 0.0f;
  }
  __syncthreads();

  const int lane = tid & 31;
  const int wave = tid >> 5;
  if (full)
    gemm1_body<true>(xs, ws, LDK, b1, h, m_base, lane, wave);
  else
    gemm1_body<false>(xs, ws, LDK, b1, h, m_base, lane, wave);
}

// ---------------------------------------------------------------------------
// Per-channel sum / sum-of-squares (BatchNorm training stats)
// ---------------------------------------------------------------------------
__global__ __launch_bounds__(256) void stats_kernel(const float* __restrict__ h,
                                                    float* __restrict__ stats) {
  const int c = threadIdx.x & (HID - 1);
  int r = blockIdx.x * 2 + (threadIdx.x >> 7);
  const int stride = gridDim.x * 2;
  float s = 0.0f, s2 = 0.0f;
  for (; r < N_NODES; r += stride) {
    float v = h[(size_t)r * HID + c];
    s += v;
    s2 += v * v;
  }
  atomicAdd(&stats[c], s);
  atomicAdd(&stats[HID + c], s2);
}

__global__ void bn_prep_kernel(const float* __restrict__ stats,
                               const float* __restrict__ g,
                               const float* __restrict__ be,
                               float* __restrict__ ss) {
  const int c = threadIdx.x;  // 128 threads
  const float inv_n = 1.0f / (float)N_NODES;
  float mu = stats[c] * inv_n;
  float var = stats[HID + c] * inv_n - mu * mu;  // biased variance
  float sc = g[c] * rsqrtf(var + BN_EPS);
  ss[c] = sc;
  ss[HID + c] = be[c] - mu * sc;
}

template <bool RELU>
__global__ __launch_bounds__(256) void bn_apply_kernel(const float* __restrict__ h,
                                                       const float* __restrict__ ss,
                                                       float* __restrict__ out) {
  const int i4 = blockIdx.x * 256 + threadIdx.x;  // float4 index, 3.2M total
  if (i4 >= N_NODES * HID / 4) return;
  const int c0 = (i4 * 4) & (HID - 1);
  float4 v = reinterpret_cast<const float4*>(h)[i4];
  float r[4] = {v.x, v.y, v.z, v.w};
#pragma unroll
  for (int j = 0; j < 4; ++j) {
    float t = r[j] * ss[c0 + j] + ss[HID + c0 + j];
    if (RELU) t = fmaxf(t, 0.0f);
    r[j] = t;
  }
  reinterpret_cast<float4*>(out)[i4] = float4{r[0], r[1], r[2], r[3]};
}

// ---------------------------------------------------------------------------
// Edge phase: in-degree count, reciprocal, scatter-add of feat[src] -> summed[dst]
// ---------------------------------------------------------------------------
__global__ __launch_bounds__(256) void count_kernel(const long long* __restrict__ ei,
                                                    float* __restrict__ cnt) {
  int e = blockIdx.x * 256 + threadIdx.x;
  if (e >= N_EDGES) return;
  atomicAdd(&cnt[(int)ei[N_EDGES + e]], 1.0f);
}

__global__ void recip_kernel(const float* __restrict__ cnt, float* __restrict__ invc) {
  int n = blockIdx.x * 256 + threadIdx.x;
  if (n < N_NODES) invc[n] = 1.0f / fmaxf(cnt[n], 1.0f);
}

// One wave32 per edge; lane owns 4 consecutive channels (coalesced 512B row).
__global__ __launch_bounds__(256) void scatter_kernel(const long long* __restrict__ ei,
                                                      const float* __restrict__ feat,
                                                      float* __restrict__ summed) {
  const int lane = threadIdx.x & 31;
  const int e = blockIdx.x * 8 + (threadIdx.x >> 5);
  if (e >= N_EDGES) return;
  const int src = (int)ei[e];
  const int dst = (int)ei[N_EDGES + e];
  const float4 v =
      *reinterpret_cast<const float4*>(&feat[(size_t)src * HID + lane * 4]);
  float* p = &summed[(size_t)dst * HID + lane * 4];
  atomicAdd(p + 0, v.x);
  atomicAdd(p + 1, v.y);
  atomicAdd(p + 2, v.z);
  atomicAdd(p + 3, v.w);
}

// W2 = [Wl | Wr] concatenated along K: [128, 256]
__global__ void build_w2_kernel(const float* __restrict__ Wl,
                                const float* __restrict__ Wr,
                                float* __restrict__ W2) {
  int i = blockIdx.x * 256 + threadIdx.x;
  if (i >= HID * 256) return;
  int n = i >> 8;
  int k = i & 255;
  W2[i] = (k < HID) ? Wl[n * HID + k] : Wr[n * HID + (k - HID)];
}

// ---------------------------------------------------------------------------
// GEMM2: sage = [agg | feat] @ W2^T + bl    (agg = summed * invc, on the fly)
// K = 256, blocked by 64; both z-block and W2-block staged in LDS so the inner
// loop is pure ds_load + v_wmma. 8 v8f accumulators per wave live across blocks.
// ---------------------------------------------------------------------------
__global__ __launch_bounds__(256) void gemm2_kernel(const float* __restrict__ summed,
                                                    const float* __restrict__ invc,
                                                    const float* __restrict__ feat,
                                                    const float* __restrict__ W2,
                                                    const float* __restrict__ bl,
                                                    float* __restrict__ sage) {
  constexpr int KB = 64;
  constexpr int LDZ = KB + 3;  // 67: odd stride -> conflict-free frag reads
  __shared__ float zs[128 * LDZ];   // ~34.3 KB
  __shared__ float wsb[128 * LDZ];  // ~34.3 KB  (W2 k-slice)
  const int tid = threadIdx.x;
  const int m_base = blockIdx.x * 128;
  const bool fullwg = (m_base + 128 <= N_NODES);  // uniform per workgroup
  const int lane = tid & 31;
  const int wave = tid >> 5;
  const int m0 = wave * 16;
  const int lm = lane & 15;
  const int lk = (lane >> 4) << 1;

  v8f acc[8];
#pragma unroll
  for (int t = 0; t < 8; ++t) acc[t] = (v8f){};

  for (int kb = 0; kb < 2 * HID; kb += KB) {
    __syncthreads();
    for (int i = tid; i < 128 * KB; i += 256) {
      int m = i >> 6;           // /KB
      int k = i & (KB - 1);
      int gk = kb + k;
      int gm = m_base + m;
      float v = 0.0f;
      if (gm < N_NODES) {
        v = (gk < HID) ? summed[(size_t)gm * HID + gk] * invc[gm]
                       : feat[(size_t)gm * HID + (gk - HID)];
      }
      zs[m * LDZ + k] = v;
      wsb[m * LDZ + k] = W2[m * (2 * HID) + gk];  // row n=m of W2, same k-slice
    }
    __syncthreads();
    const float* zrow = &zs[(m0 + lm) * LDZ + lk];
#pragma unroll
    for (int t = 0; t < 8; ++t) {
      const float* wrow = &wsb[(t * 16 + lm) * LDZ + lk];
#pragma unroll 4
      for (int k0 = 0; k0 < KB; k0 += 4) {
        v2f a, b;
        a.x = zrow[k0];  a.y = zrow[k0 + 1];
        b.x = wrow[k0];  b.y = wrow[k0 + 1];
        acc[t] = __builtin_amdgcn_wmma_f32_16x16x4_f32(false, a, false, b, (short)0,
                                                       acc[t], false, false);
      }
    }
  }

  const int rbase = m_base + m0 + ((lane >> 4) << 3);
  if (fullwg) {  // scalar fast path: unguarded stores
#pragma unroll
    for (int t = 0; t < 8; ++t) {
      const int col = t * 16 + lm;
      const float bias = bl[col];
#pragma unroll
      for (int v = 0; v < 8; ++v)
        sage[(size_t)(rbase + v) * HID + col] = acc[t][v] + bias;
    }
  } else {
#pragma unroll
    for (int t = 0; t < 8; ++t) {
      const int col = t * 16 + lm;
      const float bias = bl[col];
#pragma unroll
      for (int v = 0; v < 8; ++v) {
        int row = rbase + v;
        if (row < N_NODES) sage[(size_t)row * HID + col] = acc[t][v] + bias;
      }
    }
  }
}

// ---------------------------------------------------------------------------
extern "C" void kernel_launch(void* const* d_in, const int* in_sizes, int n_in,
                              void* d_out, int out_size, void* d_ws, size_t ws_size,
                              hipStream_t stream) {
  const float* x       = (const float*)d_in[0];
  const long long* ei  = (const long long*)d_in[1];  // int64 [2, E]
  const float* W1      = (const float*)d_in[2];
  const float* b1      = (const float*)d_in[3];
  const float* g1      = (const float*)d_in[4];
  const float* be1     = (const float*)d_in[5];
  const float* Wl      = (const float*)d_in[6];
  const float* bl      = (const float*)d_in[7];
  const float* Wr      = (const float*)d_in[8];
  const float* g2      = (const float*)d_in[9];
  const float* be2     = (const float*)d_in[10];

  const size_t NH = (size_t)N_NODES * HID;  // 12,800,000
  float* ws     = (float*)d_ws;
  float* h_buf  = ws;                 // [NH]  h, later reused as sage
  float* summed = ws + NH;            // [NH]
  float* cnt    = ws + 2 * NH;        // [N_NODES]
  float* invc   = cnt + N_NODES;      // [N_NODES]
  float* stats1 = invc + N_NODES;     // [256]
  float* stats2 = stats1 + 256;       // [256]
  float* ss1    = stats2 + 256;       // [256]
  float* ss2    = ss1 + 256;          // [256]
  float* W2     = ss2 + 256;          // [HID*256]

  float* feat_out = (float*)d_out;        // output[0]
  float* out2     = (float*)d_out + NH;   // output[1]

  hipMemsetAsync(summed, 0, NH * sizeof(float), stream);
  hipMemsetAsync(cnt, 0, N_NODES * sizeof(float), stream);
  hipMemsetAsync(stats1, 0, 512 * sizeof(float), stream);  // stats1 + stats2

  const int mblocks = (N_NODES + 127) / 128;  // 782

  build_w2_kernel<<<(HID * 256 + 255) / 256, 256, 0, stream>>>(Wl, Wr, W2);
  gemm1_kernel<<<mblocks, 256, 0, stream>>>(x, W1, b1, h_buf);
  stats_kernel<<<512, 256, 0, stream>>>(h_buf, stats1);
  bn_prep_kernel<<<1, HID, 0, stream>>>(stats1, g1, be1, ss1);
  bn_apply_kernel<true><<<(int)(NH / 4 + 255) / 256, 256, 0, stream>>>(h_buf, ss1,
                                                                       feat_out);
  count_kernel<<<(N_EDGES + 255) / 256, 256, 0, stream>>>(ei, cnt);
  recip_kernel<<<(N_NODES + 255) / 256, 256, 0, stream>>>(cnt, invc);
  scatter_kernel<<<(N_EDGES + 7) / 8, 256, 0, stream>>>(ei, feat_out, summed);
  gemm2_kernel<<<mblocks, 256, 0, stream>>>(summed, invc, feat_out, W2, bl, h_buf);
  stats_kernel<<<512, 256, 0, stream>>>(h_buf, stats2);
  bn_prep_kernel<<<1, HID, 0, stream>>>(stats2, g2, be2, ss2);
  bn_apply_kernel<false><<<(int)(NH / 4 + 255) / 256, 256, 0, stream>>>(h_buf, ss2,
                                                                        out2);
}